// AttractRepel_55465207660981
// MI455X (gfx1250) — compile-verified
//
#include <hip/hip_runtime.h>

// Problem constants (from the reference)
#define V_SZ 200000
#define D_SZ 300
#define NC4  75           // D_SZ / 4 float4 chunks per row
#define B_SZ 16384
#define L_SZ 4
#define STR  308          // LDS row stride (floats): 16B-aligned rows; m*308 mod 64 = 4*(m*13 mod 16)
                          // is a permutation over m=0..15 -> conflict-free WMMA tile reads
#define ROWS 16           // batch rows per workgroup
#define NPOOL 6           // ex_l, ex_r, ng_l, ng_r, orig_l, orig_r
#define ATTRACT_MARGIN 0.6f
#define REPEL_MARGIN   0.0f
#define REG_CONST      1e-9f
#define EPS_NORM       1e-12f

typedef __attribute__((ext_vector_type(2))) float v2f;
typedef __attribute__((ext_vector_type(8))) float v8f;

__global__ void AR_zero_out(float* out) { out[0] = 0.0f; }

// One workgroup handles 16 consecutive batch rows fully fused:
//   stage idx/len -> b128 gather+pool -> normalize -> WMMA Gram diagonals -> cost + reg -> one atomic.
__global__ __launch_bounds__(256) void AR_fused_kernel(
    const float* __restrict__ Winit, const float* __restrict__ Wdyn,
    const int* __restrict__ exl_i, const int* __restrict__ exl_n,
    const int* __restrict__ exr_i, const int* __restrict__ exr_n,
    const int* __restrict__ ngl_i, const int* __restrict__ ngl_n,
    const int* __restrict__ ngr_i, const int* __restrict__ ngr_n,
    const int* __restrict__ syn_flag, float* __restrict__ out)
{
    extern __shared__ float smem[];
    float* pool   = smem;                               // [NPOOL][ROWS][STR]
    float* simArr = smem + NPOOL * ROWS * STR;          // [3][16]
    float* accum  = simArr + 48;                        // 1 float
    int*   idxS   = (int*)(simArr + 52);                // [4][ROWS][L_SZ] = 256 ints (16B aligned)
    int*   lenI   = idxS + 4 * ROWS * L_SZ;             // [4][ROWS] = 64 ints

    const int tid  = threadIdx.x;
    const int lane = tid & 31;
    const int wave = tid >> 5;
    const int b0   = blockIdx.x * ROWS;

    if (tid == 0) accum[0] = 0.0f;

    const int* idxTab[4] = { exl_i, exr_i, ngl_i, ngr_i };
    const int* lenTab[4] = { exl_n, exr_n, ngl_n, ngr_n };

    // ---- Phase 0: stage indices & lengths into LDS (one global read each) ----
    {
        const int set = tid >> 6;          // 0..3 (tid = set*64 + r*4 + j)
        const int r   = (tid >> 2) & 15;
        const int j   = tid & 3;
        idxS[tid] = idxTab[set][(b0 + r) * L_SZ + j];
        if (tid < 64) {
            const int s2 = tid >> 4, r2 = tid & 15;
            lenI[tid] = lenTab[s2][b0 + r2];
        }
    }
    __syncthreads();

    // ---- Phase 1: masked-mean pooling with b128 gathers (the HBM-bound part) ----
    for (int e = tid; e < NPOOL * ROWS * NC4; e += 256) {
        const int t   = e / (ROWS * NC4);
        const int rem = e - t * (ROWS * NC4);
        const int r   = rem / NC4;
        const int d   = (rem - r * NC4) * 4;
        const float* table = (t < 4) ? Wdyn : Winit;
        const int set = (t < 4) ? t : (t - 4);
        const int4 ii = *(const int4*)(idxS + (set * ROWS + r) * L_SZ);  // 16B-aligned
        const int  n  = lenI[set * ROWS + r];
        const int  iv[4] = { ii.x, ii.y, ii.z, ii.w };
        float sx = 0.f, sy = 0.f, sz = 0.f, sw = 0.f;
#pragma unroll
        for (int j = 0; j < L_SZ; ++j) {
            if (j < n) {
                const float4 v = *(const float4*)(table + (long)iv[j] * D_SZ + d);
                sx += v.x; sy += v.y; sz += v.z; sw += v.w;
            }
        }
        const float fn = (float)n;
        float4 o; o.x = sx / fn; o.y = sy / fn; o.z = sz / fn; o.w = sw / fn;
        *(float4*)(pool + t * (ROWS * STR) + r * STR + d) = o;   // b128 LDS store
    }
    __syncthreads();

    // ---- Phase 2: L2-normalize the 4 dynamic pools (wave-parallel rows, b128 LDS) ----
    for (int c = wave; c < 4 * ROWS; c += 8) {
        const int t = c >> 4, r = c & 15;
        float* row = pool + t * (ROWS * STR) + r * STR;
        float s = 0.0f;
        for (int c4 = lane; c4 < NC4; c4 += 32) {
            const float4 v = *(const float4*)(row + 4 * c4);
            s += v.x * v.x + v.y * v.y + v.z * v.z + v.w * v.w;
        }
        for (int o = 16; o > 0; o >>= 1) s += __shfl_xor(s, o, 32);
        const float inv = 1.0f / fmaxf(sqrtf(s), EPS_NORM);
        for (int c4 = lane; c4 < NC4; c4 += 32) {
            float4 v = *(const float4*)(row + 4 * c4);
            v.x *= inv; v.y *= inv; v.z *= inv; v.w *= inv;
            *(float4*)(row + 4 * c4) = v;
        }
    }
    __syncthreads();

    // ---- Phase 3: three 16x16 Gram matrices via V_WMMA_F32_16X16X4_F32; keep diagonals ----
    // A 16x4 f32 layout: lanes 0-15 hold K=0,1 ; lanes 16-31 hold K=2,3 (ISA 7.12.2); B mirrors it.
    if (wave < 3) {
        const int xa[3] = { 0, 0, 1 };   // sim=<ex_l,ex_r>, sim_nl=<ex_l,ng_l>, sim_nr=<ex_r,ng_r>
        const int ya[3] = { 1, 2, 3 };
        const float* X = pool + xa[wave] * (ROWS * STR);
        const float* Y = pool + ya[wave] * (ROWS * STR);
        const int m    = lane & 15;
        const int koff = (lane >> 4) ? 2 : 0;
        v8f acc = { 0.f, 0.f, 0.f, 0.f, 0.f, 0.f, 0.f, 0.f };
        for (int k = 0; k < D_SZ; k += 4) {
            const int kk = k + koff;
            v2f A, Bv;
            A[0]  = X[m * STR + kk];  A[1]  = X[m * STR + kk + 1];
            Bv[0] = Y[m * STR + kk];  Bv[1] = Y[m * STR + kk + 1];
            acc = __builtin_amdgcn_wmma_f32_16x16x4_f32(
                false, A, false, Bv, (short)0, acc, false, false);
        }
        // Diagonal: lanes 0-7 -> M=lane in acc[lane]; lanes 24-31 -> M=lane-16 in acc[lane-24].
        const bool valid = (lane < 8) || (lane >= 24);
        const int  mm    = (lane < 8) ? lane : (lane - 16);
        const int  rsel  = (lane < 8) ? lane : (lane - 24);
        float diag = acc[0];
#pragma unroll
        for (int rr = 1; rr < 8; ++rr) diag = (rsel == rr) ? acc[rr] : diag;
        if (valid) simArr[wave * 16 + mm] = diag;
    }
    __syncthreads();

    // ---- Phase 4: hinge cost + regularizer, block reduction, one global atomic ----
    const int flag = syn_flag[0];
    float part = 0.0f;
    if (tid < ROWS) {
        const float s  = simArr[tid];
        const float nl = simArr[16 + tid];
        const float nr = simArr[32 + tid];
        if (flag == 0)
            part += fmaxf(ATTRACT_MARGIN + nl - s, 0.0f) + fmaxf(ATTRACT_MARGIN + nr - s, 0.0f);
        else
            part += fmaxf(REPEL_MARGIN - nl + s, 0.0f) + fmaxf(REPEL_MARGIN - nr + s, 0.0f);
    }
    // reg: 0.5*||ex_l - orig_l||^2 + 0.5*||ex_r - orig_r||^2, scaled by REG_CONST*B
    float dsum = 0.0f;
    for (int e = tid; e < 2 * ROWS * NC4; e += 256) {
        const int a   = e / (ROWS * NC4);
        const int rem = e - a * (ROWS * NC4);
        const int r   = rem / NC4;
        const int d   = (rem - r * NC4) * 4;
        const float4 u = *(const float4*)(pool + a * (ROWS * STR) + r * STR + d);
        const float4 w = *(const float4*)(pool + (4 + a) * (ROWS * STR) + r * STR + d);
        const float dx = u.x - w.x, dy = u.y - w.y, dz = u.z - w.z, dw = u.w - w.w;
        dsum += dx * dx + dy * dy + dz * dz + dw * dw;
    }
    part += dsum * (0.5f * REG_CONST * (float)B_SZ);

    atomicAdd(accum, part);                  // ds_add_f32
    __syncthreads();
    if (tid == 0) atomicAdd(out, accum[0]);  // global_atomic_add_f32
}

extern "C" void kernel_launch(void* const* d_in, const int* in_sizes, int n_in,
                              void* d_out, int out_size, void* d_ws, size_t ws_size,
                              hipStream_t stream) {
    const float* Winit = (const float*)d_in[0];
    const float* Wdyn  = (const float*)d_in[1];
    const int* exl_i   = (const int*)d_in[2];
    const int* exl_n   = (const int*)d_in[3];
    const int* exr_i   = (const int*)d_in[4];
    const int* exr_n   = (const int*)d_in[5];
    const int* ngl_i   = (const int*)d_in[6];
    const int* ngl_n   = (const int*)d_in[7];
    const int* ngr_i   = (const int*)d_in[8];
    const int* ngr_n   = (const int*)d_in[9];
    const int* syn     = (const int*)d_in[10];
    float* out = (float*)d_out;

    AR_zero_out<<<1, 1, 0, stream>>>(out);

    // pool + simArr(48) + accum/pad(4) + idxS(256 ints) + lenI(64 ints)
    const size_t smem_bytes =
        (size_t)(NPOOL * ROWS * STR + 52 + 256 + 64) * sizeof(float);
    AR_fused_kernel<<<B_SZ / ROWS, 256, smem_bytes, stream>>>(
        Winit, Wdyn, exl_i, exl_n, exr_i, exr_n,
        ngl_i, ngl_n, ngr_i, ngr_n, syn, out);
}